// forward_warp_89919435309451
// MI455X (gfx1250) — compile-verified
//
#include <hip/hip_runtime.h>

// Problem shape (fixed by reference setup_inputs): im0 [8,3,1080,1920] f32,
// flow [8,1080,1920,2] f32, out [8,3,1080,1920] f32.
#define BB 8
#define CC 3
#define HH 1080
#define WW 1920

constexpr int TILE = 32;                 // tile is 32x32 pixels
constexpr int HALO = 8;                  // LDS accumulator halo (|flow|<=8 stays local)
constexpr int REG  = TILE + 2 * HALO;    // 48
constexpr int REG2 = REG * REG;          // 2304 cells per channel

#ifndef USE_ASYNC_LDS
#if defined(__has_builtin)
#if __has_builtin(__builtin_amdgcn_global_load_async_to_lds_b128) && \
    __has_builtin(__builtin_amdgcn_s_wait_asynccnt)
#define USE_ASYNC_LDS 1
#endif
#endif
#endif
#ifndef USE_ASYNC_LDS
#define USE_ASYNC_LDS 0
#endif

typedef int v4i __attribute__((ext_vector_type(4)));
typedef __attribute__((address_space(1))) v4i* gptr_v4i;  // global (AS1)
typedef __attribute__((address_space(3))) v4i* lptr_v4i;  // LDS (AS3)

__device__ __forceinline__ void copy16_to_lds(const float* gp, float* lp) {
#if USE_ASYNC_LDS
  // gfx1250 async tensor-path: GLOBAL_LOAD_ASYNC_TO_LDS_B128 (ASYNCcnt tracked)
  __builtin_amdgcn_global_load_async_to_lds_b128(
      (gptr_v4i)(void*)gp, (lptr_v4i)(void*)lp,
      /*imm offset*/ 0, /*cpol*/ 0);
#else
  *(float4*)lp = *(const float4*)gp;
#endif
}

__global__ void __launch_bounds__(256)
zero_out_kernel(float4* __restrict__ out4, int n4) {
  int i = blockIdx.x * blockDim.x + threadIdx.x;
  const int stride = gridDim.x * blockDim.x;
  const float4 z = make_float4(0.f, 0.f, 0.f, 0.f);
  for (; i < n4; i += stride) out4[i] = z;
}

__global__ void __launch_bounds__(256)
splat_kernel(const float* __restrict__ im, const float* __restrict__ flow,
             float* __restrict__ out) {
  __shared__ __align__(16) float acc[CC * REG2];          // 27648 B accumulator
  __shared__ __align__(16) float ims[CC * TILE * TILE];   // 12288 B image tile

  const int tid   = threadIdx.x;
  const int tileX = blockIdx.x * TILE;
  const int tileY = blockIdx.y * TILE;
  const int b     = blockIdx.z;

  // ---- 1) kick off async stage of the 32x32x3 image tile into LDS ----------
  {
    const int row = tid >> 3;         // 0..31
    const int xv  = (tid & 7) << 2;   // 0,4,...,28  (float4 granularity)
    const int gy  = tileY + row;
    if (gy < HH) {
      const int gx = tileX + xv;      // W=1920 is a multiple of 32: always in range
#pragma unroll
      for (int c = 0; c < CC; ++c) {
        const float* gp = im + (((size_t)(b * CC + c) * HH + gy) * WW + gx);
        float* lp = &ims[(c * TILE + row) * TILE + xv];
        copy16_to_lds(gp, lp);
      }
    }
  }

  // ---- 2) zero the LDS accumulator (overlaps with the async loads) ---------
  {
    float4* a4 = (float4*)acc;
    const float4 z = make_float4(0.f, 0.f, 0.f, 0.f);
#pragma unroll
    for (int i = 0; i < (CC * REG2 / 4) / 256 + 1; ++i) {
      int j = tid + i * 256;
      if (j < CC * REG2 / 4) a4[j] = z;
    }
  }
#if USE_ASYNC_LDS
  __builtin_amdgcn_s_wait_asynccnt(0);
#endif
  __syncthreads();

  // ---- 3) splat: 4 pixels per thread, rows 8 apart (wave32 = one tile row) -
  const int px  = tid & (TILE - 1);
  const int py0 = tid >> 5;
  const float2* fl = (const float2*)flow;

#pragma unroll
  for (int k = 0; k < 4; ++k) {
    const int ty = py0 + 8 * k;
    const int gy = tileY + ty;
    if (gy >= HH) continue;
    const int gx = tileX + px;

    const float2 f = fl[(size_t)(b * HH + gy) * WW + gx];
    const float xx = (float)gx + f.x;
    const float yy = (float)gy + f.y;
    const float x0f = floorf(xx), y0f = floorf(yy);
    const int   x0  = (int)x0f,  y0  = (int)y0f;
    const float ax = xx - x0f,          ay = yy - y0f;           // (x-x0), (y-y0)
    const float bx = x0f + 1.0f - xx,   by = y0f + 1.0f - yy;    // (x1-x), (y1-y)

    const float v0 = ims[(0 * TILE + ty) * TILE + px];
    const float v1 = ims[(1 * TILE + ty) * TILE + px];
    const float v2 = ims[(2 * TILE + ty) * TILE + px];

    const int lx0 = x0 - (tileX - HALO);
    const int ly0 = y0 - (tileY - HALO);

    auto corner = [&](int cx, int cy, int lx, int ly, float w) {
      // reference drops out-of-image corners (zero weight) -> skip them
      if ((unsigned)cx >= (unsigned)WW || (unsigned)cy >= (unsigned)HH) return;
      if ((unsigned)lx < (unsigned)REG && (unsigned)ly < (unsigned)REG) {
        float* p = &acc[ly * REG + lx];            // LDS privatized accumulate
        unsafeAtomicAdd(p + 0 * REG2, v0 * w);
        unsafeAtomicAdd(p + 1 * REG2, v1 * w);
        unsafeAtomicAdd(p + 2 * REG2, v2 * w);
      } else {                                     // rare long-range splat
        float* p = out + (((size_t)(b * CC) * HH + cy) * WW + cx);
        unsafeAtomicAdd(p,                          v0 * w);
        unsafeAtomicAdd(p + (size_t)HH * WW,        v1 * w);
        unsafeAtomicAdd(p + 2 * (size_t)HH * WW,    v2 * w);
      }
    };
    corner(x0,     y0,     lx0,     ly0,     bx * by);  // NW
    corner(x0 + 1, y0,     lx0 + 1, ly0,     ax * by);  // NE
    corner(x0,     y0 + 1, lx0,     ly0 + 1, bx * ay);  // SW
    corner(x0 + 1, y0 + 1, lx0 + 1, ly0 + 1, ax * ay);  // SE
  }

  // ---- 4) flush LDS accumulator to global with atomics (skip zeros) --------
  __syncthreads();
  for (int i = tid; i < CC * REG2; i += 256) {
    const float v = acc[i];
    if (v != 0.0f) {
      const int c  = i / REG2;
      const int r  = i - c * REG2;
      const int ly = r / REG;
      const int lx = r - ly * REG;
      const int gy = tileY - HALO + ly;
      const int gx = tileX - HALO + lx;
      if ((unsigned)gx < (unsigned)WW && (unsigned)gy < (unsigned)HH) {
        unsafeAtomicAdd(out + (((size_t)(b * CC + c) * HH + gy) * WW + gx), v);
      }
    }
  }
}

extern "C" void kernel_launch(void* const* d_in, const int* in_sizes, int n_in,
                              void* d_out, int out_size, void* d_ws, size_t ws_size,
                              hipStream_t stream) {
  const float* im   = (const float*)d_in[0];   // [8,3,1080,1920]
  const float* flow = (const float*)d_in[1];   // [8,1080,1920,2]
  float* out = (float*)d_out;                  // [8,3,1080,1920]

  // zero-init output (harness poisons it); out_size = 49,766,400 = 0 mod 4
  const int n4 = out_size / 4;
  zero_out_kernel<<<4096, 256, 0, stream>>>((float4*)out, n4);

  dim3 grid(WW / TILE, (HH + TILE - 1) / TILE, BB);  // 60 x 34 x 8
  splat_kernel<<<grid, 256, 0, stream>>>(im, flow, out);
}